// PromptGCN_30983894073822
// MI455X (gfx1250) — compile-verified
//
#include <hip/hip_runtime.h>
#include <hip/hip_bf16.h>
#include <math.h>
#include <stdint.h>

// PromptGCN bidirectional aggregation for MI455X (gfx1250), CSR-based (atomic-free
// feature aggregation). No dense matmul in this op -> WMMA not applicable; CDNA5
// paths used: async global->LDS copy engine (ASYNCcnt), ds-permute index
// broadcast, global prefetch.

#define N_USERS 100000
#define N_ITEMS 50000
#define N_EDGES 2000000
#define D 64
#define ALPHA 0.5f
#define NORM_2 -1.0f
#define TILE 256   // edges staged per workgroup in count/bucket passes

__global__ void zero_int_kernel(int* __restrict__ p, int n) {
  int i = blockIdx.x * blockDim.x + threadIdx.x;
  int stride = gridDim.x * blockDim.x;
  for (; i < n; i += stride) p[i] = 0;
}

// Stage a TILE of src/dst indices into LDS with the CDNA5 async copy engine,
// then histogram both endpoints (int atomics).
__global__ void count_kernel(const int* __restrict__ src, const int* __restrict__ dst,
                             int* __restrict__ cnt_src, int* __restrict__ cnt_dst) {
  __shared__ int s_src[TILE];
  __shared__ int s_dst[TILE];
  const int t = threadIdx.x;
  const long long e = (long long)blockIdx.x * TILE + t;
  if (e < (long long)N_EDGES) {
    unsigned ls  = (unsigned)(uintptr_t)(&s_src[t]);
    unsigned ld  = (unsigned)(uintptr_t)(&s_dst[t]);
    unsigned off = (unsigned)(e * 4ll);
    asm volatile("global_load_async_to_lds_b32 %0, %1, %2"
                 :: "v"(ls), "v"(off), "s"(src) : "memory");
    asm volatile("global_load_async_to_lds_b32 %0, %1, %2"
                 :: "v"(ld), "v"(off), "s"(dst) : "memory");
  }
  asm volatile("s_wait_asynccnt 0" ::: "memory");
  if (e < (long long)N_EDGES) {
    atomicAdd(&cnt_src[s_src[t]], 1);
    atomicAdd(&cnt_dst[s_dst[t]], 1);
  }
}

// Single-workgroup exclusive scan over up to ~100K bins.
// Writes off[0..n] (off[n] = total) and a working copy cur[0..n-1].
__global__ void exscan_kernel(const int* __restrict__ cnt, int* __restrict__ off,
                              int* __restrict__ cur, int n) {
  __shared__ int sums[1024];
  const int t = threadIdx.x;
  const int chunk = (n + 1023) / 1024;
  const int lo = t * chunk;
  const int hi = (lo + chunk < n) ? (lo + chunk) : n;
  int s = 0;
  for (int i = lo; i < hi; ++i) s += cnt[i];
  sums[t] = s;
  __syncthreads();
  for (int d = 1; d < 1024; d <<= 1) {     // inclusive Hillis-Steele scan
    int v = (t >= d) ? sums[t - d] : 0;
    __syncthreads();
    sums[t] += v;
    __syncthreads();
  }
  int run = (t == 0) ? 0 : sums[t - 1];
  for (int i = lo; i < hi; ++i) {
    off[i] = run;
    cur[i] = run;
    run += cnt[i];
  }
  if (t == 1023) off[n] = sums[1023];
}

// Bucket edges into both CSR orderings:
//   by_dst_src: src endpoints grouped by dst   (forward gather list)
//   by_src_dst: dst endpoints grouped by src   (backward gather list)
__global__ void bucket_kernel(const int* __restrict__ src, const int* __restrict__ dst,
                              int* __restrict__ cur_dst, int* __restrict__ cur_src,
                              int* __restrict__ by_dst_src, int* __restrict__ by_src_dst) {
  __shared__ int s_src[TILE];
  __shared__ int s_dst[TILE];
  const int t = threadIdx.x;
  const long long e = (long long)blockIdx.x * TILE + t;
  if (e < (long long)N_EDGES) {
    unsigned ls  = (unsigned)(uintptr_t)(&s_src[t]);
    unsigned ld  = (unsigned)(uintptr_t)(&s_dst[t]);
    unsigned off = (unsigned)(e * 4ll);
    asm volatile("global_load_async_to_lds_b32 %0, %1, %2"
                 :: "v"(ls), "v"(off), "s"(src) : "memory");
    asm volatile("global_load_async_to_lds_b32 %0, %1, %2"
                 :: "v"(ld), "v"(off), "s"(dst) : "memory");
  }
  asm volatile("s_wait_asynccnt 0" ::: "memory");
  if (e < (long long)N_EDGES) {
    const int s = s_src[t];
    const int d = s_dst[t];
    const int p = atomicAdd(&cur_dst[d], 1);
    by_dst_src[p] = s;
    const int q = atomicAdd(&cur_src[s], 1);
    by_src_dst[q] = d;
  }
}

// softmax(edge_w, axis=0) statistics over N_ITEMS scalars: sm[0]=max, sm[1]=sum(exp(x-max))
__global__ void softmax_stats_kernel(const float* __restrict__ ew, float* __restrict__ sm) {
  __shared__ float red[1024];
  const int t = threadIdx.x;
  float m = -INFINITY;
  for (int i = t; i < N_ITEMS; i += 1024) m = fmaxf(m, ew[i]);
  red[t] = m;
  __syncthreads();
  for (int s = 512; s > 0; s >>= 1) {
    if (t < s) red[t] = fmaxf(red[t], red[t + s]);
    __syncthreads();
  }
  const float mx = red[0];
  __syncthreads();
  float acc = 0.0f;
  for (int i = t; i < N_ITEMS; i += 1024) acc += expf(ew[i] - mx);
  red[t] = acc;
  __syncthreads();
  for (int s = 512; s > 0; s >>= 1) {
    if (t < s) red[t] += red[t + s];
    __syncthreads();
  }
  if (t == 0) { sm[0] = mx; sm[1] = red[0]; }
}

// Forward aggregation fused with item transform. 16 lanes per item; each lane
// owns a float4 chunk of the 64-float row. Neighbor indices loaded once per
// 16-lane group and broadcast with __shfl (ds-permute HW).
// rst[i] = (sum_{s in N(i)} h_user[s] + ALPHA*tanh(pf[i])) * softmax_w(i) / max(deg,1)
__global__ void item_agg_kernel(const float* __restrict__ h_user,
                                const int* __restrict__ by_dst_src,
                                const int* __restrict__ off_dst,
                                const float* __restrict__ pf,
                                const float* __restrict__ ew,
                                const float* __restrict__ sm,
                                float* __restrict__ rst) {
  const int t = threadIdx.x;
  const int sub = t & 15;
  const int item = blockIdx.x * (blockDim.x >> 4) + (t >> 4);
  if (item >= N_ITEMS) return;
  const int begin = off_dst[item];
  const int end = off_dst[item + 1];
  float4 acc = make_float4(0.f, 0.f, 0.f, 0.f);
  for (int e0 = begin; e0 < end; e0 += 16) {
    if (e0 + 16 < end) __builtin_prefetch(&by_dst_src[e0 + 16], 0, 1);
    int myidx = 0;
    if (e0 + sub < end) myidx = by_dst_src[e0 + sub];
    const int cnt = (end - e0 < 16) ? (end - e0) : 16;
    for (int j = 0; j < cnt; ++j) {
      const int s = __shfl(myidx, j, 16);
      const float4 v = ((const float4*)(h_user + (size_t)s * D))[sub];
      acc.x += v.x; acc.y += v.y; acc.z += v.z; acc.w += v.w;
    }
  }
  const float w = expf(ew[item] - sm[0]) / sm[1];
  const float scale = w / fmaxf((float)(end - begin), 1.0f);
  const float4 p = ((const float4*)pf)[item * (D / 4) + sub];
  float4 r;
  r.x = (acc.x + ALPHA * tanhf(p.x)) * scale;
  r.y = (acc.y + ALPHA * tanhf(p.y)) * scale;
  r.z = (acc.z + ALPHA * tanhf(p.z)) * scale;
  r.w = (acc.w + ALPHA * tanhf(p.w)) * scale;
  ((float4*)rst)[item * (D / 4) + sub] = r;
}

// Backward aggregation fused with degree power norm.
// out[u] = (sum_{d in N(u)} rst[d]) * pow(max(deg,1), NORM_2)
__global__ void user_agg_kernel(const float* __restrict__ rst,
                                const int* __restrict__ by_src_dst,
                                const int* __restrict__ off_src,
                                float* __restrict__ out) {
  const int t = threadIdx.x;
  const int sub = t & 15;
  const int user = blockIdx.x * (blockDim.x >> 4) + (t >> 4);
  if (user >= N_USERS) return;
  const int begin = off_src[user];
  const int end = off_src[user + 1];
  float4 acc = make_float4(0.f, 0.f, 0.f, 0.f);
  for (int e0 = begin; e0 < end; e0 += 16) {
    if (e0 + 16 < end) __builtin_prefetch(&by_src_dst[e0 + 16], 0, 1);
    int myidx = 0;
    if (e0 + sub < end) myidx = by_src_dst[e0 + sub];
    const int cnt = (end - e0 < 16) ? (end - e0) : 16;
    for (int j = 0; j < cnt; ++j) {
      const int d = __shfl(myidx, j, 16);
      const float4 v = ((const float4*)(rst + (size_t)d * D))[sub];
      acc.x += v.x; acc.y += v.y; acc.z += v.z; acc.w += v.w;
    }
  }
  const float s = powf(fmaxf((float)(end - begin), 1.0f), NORM_2);
  float4 r;
  r.x = acc.x * s; r.y = acc.y * s; r.z = acc.z * s; r.w = acc.w * s;
  ((float4*)out)[user * (D / 4) + sub] = r;
}

extern "C" void kernel_launch(void* const* d_in, const int* in_sizes, int n_in,
                              void* d_out, int out_size, void* d_ws, size_t ws_size,
                              hipStream_t stream) {
  (void)in_sizes; (void)n_in; (void)out_size; (void)ws_size;

  const float* h_user = (const float*)d_in[0];  // [N_USERS, D]
  const float* pf     = (const float*)d_in[1];  // [N_ITEMS, D]
  const float* ew     = (const float*)d_in[2];  // [N_ITEMS, 1]
  const int*   src    = (const int*)d_in[3];    // [N_EDGES]
  const int*   dst    = (const int*)d_in[4];    // [N_EDGES]
  float* out = (float*)d_out;                   // [N_USERS, D]

  // Workspace carve-up (all 4-byte types; rst first for 16B alignment).
  float* rst        = (float*)d_ws;                          // N_ITEMS*D
  float* sm         = rst + (size_t)N_ITEMS * D;             // 2
  int*   cnt_dst    = (int*)(sm + 2);                        // N_ITEMS
  int*   off_dst    = cnt_dst + N_ITEMS;                     // N_ITEMS+1
  int*   cur_dst    = off_dst + (N_ITEMS + 1);               // N_ITEMS
  int*   cnt_src    = cur_dst + N_ITEMS;                     // N_USERS
  int*   off_src    = cnt_src + N_USERS;                     // N_USERS+1
  int*   cur_src    = off_src + (N_USERS + 1);               // N_USERS
  int*   by_dst_src = cur_src + N_USERS;                     // N_EDGES
  int*   by_src_dst = by_dst_src + N_EDGES;                  // N_EDGES
  // total ~30.6 MB

  // 1) reset histograms (contiguous: cnt_dst..cur_src region only needs cnt zeroed)
  zero_int_kernel<<<512, 256, 0, stream>>>(cnt_dst, N_ITEMS);
  zero_int_kernel<<<512, 256, 0, stream>>>(cnt_src, N_USERS);

  // 2) degree histograms (async-LDS staged edge tiles)
  const int egrid = (N_EDGES + TILE - 1) / TILE;
  count_kernel<<<egrid, TILE, 0, stream>>>(src, dst, cnt_src, cnt_dst);

  // 3) exclusive scans -> CSR offsets + cursors
  exscan_kernel<<<1, 1024, 0, stream>>>(cnt_dst, off_dst, cur_dst, N_ITEMS);
  exscan_kernel<<<1, 1024, 0, stream>>>(cnt_src, off_src, cur_src, N_USERS);

  // 4) bucket both CSR orderings
  bucket_kernel<<<egrid, TILE, 0, stream>>>(src, dst, cur_dst, cur_src,
                                            by_dst_src, by_src_dst);

  // 5) node-softmax stats
  softmax_stats_kernel<<<1, 1024, 0, stream>>>(ew, sm);

  // 6) forward aggregation fused with tanh/softmax/deg transform (no atomics)
  const int igrid = (N_ITEMS + 15) / 16;   // 16 items per 256-thread block
  item_agg_kernel<<<igrid, 256, 0, stream>>>(h_user, by_dst_src, off_dst,
                                             pf, ew, sm, rst);

  // 7) backward aggregation fused with deg^NORM_2 (no atomics, out fully written)
  const int ugrid = (N_USERS + 15) / 16;
  user_agg_kernel<<<ugrid, 256, 0, stream>>>(rst, by_src_dst, off_src, out);
}